// AttentionalFM_15272903705281
// MI455X (gfx1250) — compile-verified
//
#include <hip/hip_runtime.h>
#include <math.h>

#define NFIELDS 50
#define EMB_D   64
#define ATT_D   32
#define PAIRS   1225          // NFIELDS*(NFIELDS-1)/2
#define NTILES  77            // ceil(PAIRS/16)
#define THREADS 256
#define NWAVES  (THREADS/32)
#define ROWPAD  65            // 64 + 1 to break LDS bank conflicts

typedef __attribute__((ext_vector_type(2))) float v2f;
typedef __attribute__((ext_vector_type(8))) float v8f;

__global__ __launch_bounds__(THREADS)
void afm_kernel(const int* __restrict__ features,
                const float* __restrict__ emb_table,
                const float* __restrict__ coeff_table,
                const float* __restrict__ bias,
                const float* __restrict__ att_w,
                const float* __restrict__ att_b,
                const float* __restrict__ p_vec,
                const float* __restrict__ w_vec,
                float* __restrict__ out)
{
    __shared__ float s_emb[NFIELDS][ROWPAD];     // gathered embeddings
    __shared__ float s_attw[ATT_D][ROWPAD];      // attention weights
    __shared__ float s_scores[NTILES * 16];      // scores -> exp weights
    __shared__ unsigned char s_pi[PAIRS];
    __shared__ unsigned char s_pj[PAIRS];
    __shared__ float s_red[THREADS];
    __shared__ float s_part[4][EMB_D];
    __shared__ float s_coeff[NFIELDS];
    __shared__ int   s_feat[NFIELDS];
    __shared__ float s_m, s_z;

    const int b   = blockIdx.x;
    const int tid = threadIdx.x;
    const int wave   = tid >> 5;
    const int lane   = tid & 31;
    const int lane16 = lane & 15;
    const int hi     = lane >> 4;       // 0: lanes 0-15, 1: lanes 16-31

    // ---- stage feature ids ----
    if (tid < NFIELDS) s_feat[tid] = features[(size_t)b * NFIELDS + tid];
    __syncthreads();

    // ---- stage coeffs, embeddings, att_w, pair tables ----
    if (tid < NFIELDS) s_coeff[tid] = coeff_table[s_feat[tid]];

    for (int idx = tid; idx < NFIELDS * 16; idx += THREADS) {   // 50 rows * 16 float4
        int row = idx >> 4, c = (idx & 15) * 4;
        const float4 v = *(const float4*)(emb_table + (size_t)s_feat[row] * EMB_D + c);
        s_emb[row][c + 0] = v.x; s_emb[row][c + 1] = v.y;
        s_emb[row][c + 2] = v.z; s_emb[row][c + 3] = v.w;
    }
    for (int idx = tid; idx < ATT_D * 16; idx += THREADS) {     // 32 rows * 16 float4
        int row = idx >> 4, c = (idx & 15) * 4;
        const float4 v = *(const float4*)(att_w + row * EMB_D + c);
        s_attw[row][c + 0] = v.x; s_attw[row][c + 1] = v.y;
        s_attw[row][c + 2] = v.z; s_attw[row][c + 3] = v.w;
    }
    for (int q = tid; q < PAIRS; q += THREADS) {                // triu(i<j) row-major
        int i = 0, off = 0;
        while (q >= off + (NFIELDS - 1 - i)) { off += NFIELDS - 1 - i; ++i; }
        s_pi[q] = (unsigned char)i;
        s_pj[q] = (unsigned char)(i + 1 + (q - off));
    }
    __syncthreads();

    // ---- pass 1: scores via V_WMMA_F32_16X16X4_F32 ----
    const float ab0 = att_b[lane16];
    const float ab1 = att_b[lane16 + 16];
    const float pp0 = p_vec[lane16];
    const float pp1 = p_vec[lane16 + 16];

    for (int t = wave; t < NTILES; t += NWAVES) {
        const int base = t * 16;
        const int myp  = base + lane16;          // A-matrix row M = lane16
        const bool valid = (myp < PAIRS);
        const int ri = valid ? (int)s_pi[myp] : 0;
        const int rj = valid ? (int)s_pj[myp] : 0;

        v8f c0 = {}; v8f c1 = {};
        #pragma unroll
        for (int kk = 0; kk < 16; ++kk) {
            const int k0 = kk * 4 + hi * 2;      // K base for this lane half
            v2f a;
            float e0 = s_emb[ri][k0]     * s_emb[rj][k0];
            float e1 = s_emb[ri][k0 + 1] * s_emb[rj][k0 + 1];
            a.x = valid ? e0 : 0.0f;
            a.y = valid ? e1 : 0.0f;
            v2f b0, b1;                          // B = att_w^T, N = lane16 (+16)
            b0.x = s_attw[lane16][k0];      b0.y = s_attw[lane16][k0 + 1];
            b1.x = s_attw[lane16 + 16][k0]; b1.y = s_attw[lane16 + 16][k0 + 1];
            c0 = __builtin_amdgcn_wmma_f32_16x16x4_f32(false, a, false, b0,
                                                       (short)0, c0, false, false);
            c1 = __builtin_amdgcn_wmma_f32_16x16x4_f32(false, a, false, b1,
                                                       (short)0, c1, false, false);
        }
        // score[m] = sum_a relu(h[m][a]+att_b[a]) * p[a]
        #pragma unroll
        for (int r = 0; r < 8; ++r) {
            float h0 = c0[r] + ab0; h0 = h0 > 0.0f ? h0 : 0.0f;
            float h1 = c1[r] + ab1; h1 = h1 > 0.0f ? h1 : 0.0f;
            float s = h0 * pp0 + h1 * pp1;       // partial over att units
            s += __shfl_xor(s, 1, 32);
            s += __shfl_xor(s, 2, 32);
            s += __shfl_xor(s, 4, 32);
            s += __shfl_xor(s, 8, 32);           // sum within 16-lane half
            if (lane16 == 0) {
                int pr = base + r + 8 * hi;      // M = r (lo half) / r+8 (hi half)
                if (pr < PAIRS) s_scores[pr] = s;
            }
        }
    }
    __syncthreads();

    // ---- softmax: max ----
    float lm = -INFINITY;
    for (int q = tid; q < PAIRS; q += THREADS) lm = fmaxf(lm, s_scores[q]);
    s_red[tid] = lm;
    __syncthreads();
    for (int off = THREADS / 2; off > 0; off >>= 1) {
        if (tid < off) s_red[tid] = fmaxf(s_red[tid], s_red[tid + off]);
        __syncthreads();
    }
    if (tid == 0) s_m = s_red[0];
    __syncthreads();

    // ---- softmax: sum of exp (scores -> unnormalized weights) ----
    const float m = s_m;
    float lz = 0.0f;
    for (int q = tid; q < PAIRS; q += THREADS) {
        float e = expf(s_scores[q] - m);
        s_scores[q] = e;
        lz += e;
    }
    __syncthreads();
    s_red[tid] = lz;
    __syncthreads();
    for (int off = THREADS / 2; off > 0; off >>= 1) {
        if (tid < off) s_red[tid] += s_red[tid + off];
        __syncthreads();
    }
    if (tid == 0) s_z = s_red[0];
    __syncthreads();

    // ---- pass 2: pooled[d] = sum_p w_p * emb_i[d]*emb_j[d] ----
    const int d = tid & 63;
    const int g = tid >> 6;                      // 4 groups over pairs
    float acc = 0.0f;
    for (int q = g; q < PAIRS; q += 4)
        acc += s_scores[q] * s_emb[s_pi[q]][d] * s_emb[s_pj[q]][d];
    s_part[g][d] = acc;
    __syncthreads();

    if (tid < EMB_D) {
        float pooled = s_part[0][tid] + s_part[1][tid] + s_part[2][tid] + s_part[3][tid];
        s_red[tid] = pooled * w_vec[tid];
    }
    __syncthreads();
    if (tid == 0) {
        float dot = 0.0f;
        for (int q = 0; q < EMB_D; ++q) dot += s_red[q];
        float lin = 0.0f;
        for (int f = 0; f < NFIELDS; ++f) lin += s_coeff[f];
        out[b] = dot / s_z + lin + bias[0];
    }
}

extern "C" void kernel_launch(void* const* d_in, const int* in_sizes, int n_in,
                              void* d_out, int out_size, void* d_ws, size_t ws_size,
                              hipStream_t stream) {
    const int*   features    = (const int*)d_in[0];
    const float* emb_table   = (const float*)d_in[1];
    const float* coeff_table = (const float*)d_in[2];
    const float* bias        = (const float*)d_in[3];
    const float* att_w       = (const float*)d_in[4];
    const float* att_b       = (const float*)d_in[5];
    const float* p_vec       = (const float*)d_in[6];
    const float* w_vec       = (const float*)d_in[7];
    float* outp = (float*)d_out;

    const int batch = in_sizes[0] / NFIELDS;     // 2048
    afm_kernel<<<batch, THREADS, 0, stream>>>(features, emb_table, coeff_table, bias,
                                              att_w, att_b, p_vec, w_vec, outp);
}